// constraints_enforcement_numba_83751862272556
// MI455X (gfx1250) — compile-verified
//
#include <hip/hip_runtime.h>

// ---------------------------------------------------------------------------
// Rod edge-constraint solver for MI455X (gfx1250, wave32).
//   lane == rod; sequential 127-edge sweep held in registers (3 floats carry).
//   Memory-bound: ~117 MB/call -> ~5us @ 23.3 TB/s HBM (L2-resident on replay:
//   whole working set < 192 MB L2).
//   Batch-global early-exit handled by speculation + exact fallback kernel.
// CDNA5 paths: async global->LDS triple buffering (ASYNCcnt-tracked),
//   global_prefetch_b8, s_wait_asynccnt / s_wait_tensorcnt split counters.
//   No WMMA by design: batched per-rod 3x3 matvecs, AI ~0.6 flop/byte.
// ---------------------------------------------------------------------------

#define RODS    8192
#define NVERT   128
#define NEDGE   127
#define THREADS 32                    // one wave per workgroup
#define BLOCKS  (RODS / THREADS)      // 256 workgroups -> spread over WGPs
#define NWAVES  (RODS / 32)           // 256 waves total
#define TOLF    1e-6f

// float4 with 4-byte alignment: gfx1250 DWORD alignment mode permits
// B128/B96 vector memops at DWORD alignment, so these lower to wide loads.
typedef float v4fa4 __attribute__((ext_vector_type(4), aligned(4)));

#if defined(__has_builtin)
#  if __has_builtin(__builtin_amdgcn_global_load_async_to_lds_b32) && \
      __has_builtin(__builtin_amdgcn_s_wait_asynccnt)
#    define HAVE_ASYNC_LDS 1
#  endif
#endif

#if defined(HAVE_ASYNC_LDS)
// per-lane 4-byte async copy global -> LDS (ASYNCcnt-tracked).
// Probe-verified signature: (AS1 int* src, AS3 int* dst, imm offset, imm cpol).
#define ASYNC_CP_B32(ldst, gsrc)                                            \
  __builtin_amdgcn_global_load_async_to_lds_b32(                            \
      (__attribute__((address_space(1))) int*)(gsrc),                       \
      (__attribute__((address_space(3))) int*)(ldst), 0, 0)
#endif

// ---------------------------------------------------------------------------
// Kernel 1: speculative fast path (assumes the batch-global all_small never
// fires). Records, per wave, a 127-bit mask of "all 32 lanes had |l|<TOL".
// ---------------------------------------------------------------------------
__global__ __launch_bounds__(THREADS) void rods_fast(
    const float* __restrict__ verts, const float* __restrict__ nl,
    const float* __restrict__ sc,    const float* __restrict__ ms,
    const float* __restrict__ zm,    float* __restrict__ out,
    unsigned* __restrict__ wave_masks)
{
  const int tid = threadIdx.x;
  const int b   = blockIdx.x * THREADS + tid;
  const float* vb = verts + (size_t)b * (NVERT * 3);
  float*       ob = out   + (size_t)b * (NVERT * 3);
  const size_t e0 = (size_t)b * NEDGE;            // nl / zm row
  const size_t s0 = (size_t)(2 * b) * NEDGE;      // scale row, endpoint 0
  const size_t s1 = s0 + NEDGE;                   // scale row, endpoint 1
  const float* m0p = ms + s0 * 9;                 // 3x3 blocks, endpoint 0
  const float* m1p = ms + s1 * 9;                 // 3x3 blocks, endpoint 1

  float v0x = vb[0], v0y = vb[1], v0z = vb[2];
  unsigned msk0 = 0u, msk1 = 0u, msk2 = 0u, msk3 = 0u;

#if defined(HAVE_ASYNC_LDS)
  // triple-buffered vertex staging through LDS via async copies; a restage
  // always targets a buffer last read a full iteration earlier (no WAR race
  // between ds_load and the unordered async LDS write).
  __shared__ float vstage[3][THREADS][3];
  {
    const float* p1 = vb + 3;   // vertex 1 -> buffer 0
    const float* p2 = vb + 6;   // vertex 2 -> buffer 1
    ASYNC_CP_B32(&vstage[0][tid][0], p1 + 0);
    ASYNC_CP_B32(&vstage[0][tid][1], p1 + 1);
    ASYNC_CP_B32(&vstage[0][tid][2], p1 + 2);
    ASYNC_CP_B32(&vstage[1][tid][0], p2 + 0);
    ASYNC_CP_B32(&vstage[1][tid][1], p2 + 1);
    ASYNC_CP_B32(&vstage[1][tid][2], p2 + 2);
  }
#endif

#pragma unroll 3
  for (int i = 0; i < NEDGE; ++i) {
    float v1x, v1y, v1z;
#if defined(HAVE_ASYNC_LDS)
    __builtin_amdgcn_s_wait_asynccnt(3);          // oldest 3-copy batch done
    v1x = vstage[i % 3][tid][0];
    v1y = vstage[i % 3][tid][1];
    v1z = vstage[i % 3][tid][2];
    if (i + 3 <= NVERT - 1) {                     // restage vertex i+3
      const float* pn = vb + (size_t)(i + 3) * 3;
      ASYNC_CP_B32(&vstage[(i + 2) % 3][tid][0], pn + 0);
      ASYNC_CP_B32(&vstage[(i + 2) % 3][tid][1], pn + 1);
      ASYNC_CP_B32(&vstage[(i + 2) % 3][tid][2], pn + 2);
    }
#else
    const float* vp = vb + (size_t)(i + 1) * 3;
    v1x = vp[0]; v1y = vp[1]; v1z = vp[2];
#endif
    // prefetch mass_scale two edges ahead (global_prefetch_b8)
    {
      const int ip = (i + 2 < NEDGE) ? i + 2 : NEDGE - 1;
      __builtin_prefetch(m0p + (size_t)ip * 9, 0, 1);
      __builtin_prefetch(m1p + (size_t)ip * 9, 0, 1);
    }

    const float z    = zm[e0 + i];
    const float nlv  = nl[e0 + i];
    const float nlsq = nlv * nlv;
    const float ex = (v1x - v0x) * z;
    const float ey = (v1y - v0y) * z;
    const float ez = (v1z - v0z) * z;
    const float denom = nlsq + ex * ex + ey * ey + ez * ez;
    const float l = (z != 0.0f) ? (1.0f - 2.0f * nlsq / denom) : 0.0f;

    const int wall = __all(fabsf(l) < TOLF);      // wave-uniform
    const unsigned bit = (unsigned)(wall != 0) << (i & 31);
    switch (i >> 5) {
      case 0: msk0 |= bit; break;
      case 1: msk1 |= bit; break;
      case 2: msk2 |= bit; break;
      default: msk3 |= bit; break;
    }

    const float w0 = l / sc[s0 + i];
    const float w1 = l / sc[s1 + i];
    const float* a = m0p + (size_t)i * 9;
    const float* c = m1p + (size_t)i * 9;
    // 3x3 blocks as two b128 + one b32 (DWORD-aligned wide loads)
    const v4fa4 A0 = *(const v4fa4*)(a + 0);
    const v4fa4 A1 = *(const v4fa4*)(a + 4);
    const float A8 = a[8];
    const v4fa4 C0 = *(const v4fa4*)(c + 0);
    const v4fa4 C1 = *(const v4fa4*)(c + 4);
    const float C8 = c[8];

    v0x += w0 * (A0.x * ex + A0.y * ey + A0.z * ez);
    v0y += w0 * (A0.w * ex + A1.x * ey + A1.y * ez);
    v0z += w0 * (A1.z * ex + A1.w * ey + A8   * ez);
    const float n1x = v1x + w1 * (C0.x * ex + C0.y * ey + C0.z * ez);
    const float n1y = v1y + w1 * (C0.w * ex + C1.x * ey + C1.y * ez);
    const float n1z = v1z + w1 * (C1.z * ex + C1.w * ey + C8   * ez);

    ob[(size_t)i * 3 + 0] = v0x;                  // vertex i is final now
    ob[(size_t)i * 3 + 1] = v0y;
    ob[(size_t)i * 3 + 2] = v0z;
    v0x = n1x; v0y = n1y; v0z = n1z;              // carry updated vertex i+1
  }
  ob[(size_t)(NVERT - 1) * 3 + 0] = v0x;
  ob[(size_t)(NVERT - 1) * 3 + 1] = v0y;
  ob[(size_t)(NVERT - 1) * 3 + 2] = v0z;

  if ((tid & 31) == 0) {
    const unsigned wid = (unsigned)(blockIdx.x * (THREADS / 32) + (tid >> 5));
    wave_masks[(size_t)wid * 4 + 0] = msk0;
    wave_masks[(size_t)wid * 4 + 1] = msk1;
    wave_masks[(size_t)wid * 4 + 2] = msk2;
    wave_masks[(size_t)wid * 4 + 3] = msk3;
  }
#if defined(__has_builtin)
#if __has_builtin(__builtin_amdgcn_s_wait_tensorcnt)
  __builtin_amdgcn_s_wait_tensorcnt(0);           // gfx1250 split-counter wait
#endif
#endif
}

// ---------------------------------------------------------------------------
// Kernel 2 (1 block): AND wave masks across the batch. If any step was
// all-small batch-wide, speculation may have diverged -> flag exact rerun.
// Also (re)initializes the exact kernel's barrier/flag state every call.
// ctrl[0]=need_fallback, ctrl[1]=barrier counter, ctrl[2..128]=per-step flags.
// ---------------------------------------------------------------------------
__global__ __launch_bounds__(128) void rods_reduce(
    const unsigned* __restrict__ wm, unsigned* __restrict__ ctrl)
{
  __shared__ unsigned sh[4];
  const int t = threadIdx.x;
  if (t < 4) sh[t] = 0xFFFFFFFFu;
  __syncthreads();
  unsigned a0 = 0xFFFFFFFFu, a1 = 0xFFFFFFFFu, a2 = 0xFFFFFFFFu, a3 = 0xFFFFFFFFu;
  for (int w = t; w < NWAVES; w += 128) {
    a0 &= wm[(size_t)w * 4 + 0];
    a1 &= wm[(size_t)w * 4 + 1];
    a2 &= wm[(size_t)w * 4 + 2];
    a3 &= wm[(size_t)w * 4 + 3];
  }
  atomicAnd(&sh[0], a0); atomicAnd(&sh[1], a1);
  atomicAnd(&sh[2], a2); atomicAnd(&sh[3], a3);
  __syncthreads();
  if (t < NEDGE)  ctrl[2 + t] = 0xFFFFFFFFu;      // exact per-step AND flags
  if (t == NEDGE) ctrl[1] = 0u;                   // barrier counter
  if (t == 0) {
    const unsigned any = sh[0] | sh[1] | sh[2] | (sh[3] & 0x7FFFFFFFu);
    ctrl[0] = any ? 1u : 0u;
  }
}

// ---------------------------------------------------------------------------
// Kernel 3: exact semantics (cold path). Early-exits unless flagged. Uses a
// monotonic atomic-counter grid barrier per edge step (256 resident blocks).
// ---------------------------------------------------------------------------
__global__ __launch_bounds__(THREADS) void rods_exact(
    const float* __restrict__ verts, const float* __restrict__ nl,
    const float* __restrict__ sc,    const float* __restrict__ ms,
    const float* __restrict__ zm,    float* __restrict__ out,
    unsigned* __restrict__ ctrl)
{
  if (__hip_atomic_load(&ctrl[0], __ATOMIC_ACQUIRE, __HIP_MEMORY_SCOPE_AGENT) == 0u)
    return;                                       // common case: nothing to do

  const int tid = threadIdx.x;
  const int b   = blockIdx.x * THREADS + tid;
  const float* vb = verts + (size_t)b * (NVERT * 3);
  float*       ob = out   + (size_t)b * (NVERT * 3);
  const size_t e0 = (size_t)b * NEDGE;
  const size_t s0 = (size_t)(2 * b) * NEDGE;
  const size_t s1 = s0 + NEDGE;
  const float* m0p = ms + s0 * 9;
  const float* m1p = ms + s1 * 9;

  __shared__ unsigned blkflag;
  float v0x = vb[0], v0y = vb[1], v0z = vb[2];

  for (int i = 0; i < NEDGE; ++i) {
    const float* vp = vb + (size_t)(i + 1) * 3;
    const float v1x = vp[0], v1y = vp[1], v1z = vp[2];
    const float z    = zm[e0 + i];
    const float nlv  = nl[e0 + i];
    const float nlsq = nlv * nlv;
    const float ex = (v1x - v0x) * z;
    const float ey = (v1y - v0y) * z;
    const float ez = (v1z - v0z) * z;
    const float denom = nlsq + ex * ex + ey * ey + ez * ez;
    const float l = (z != 0.0f) ? (1.0f - 2.0f * nlsq / denom) : 0.0f;
    const int wall = __all(fabsf(l) < TOLF);

    if (tid == 0) blkflag = 0xFFFFFFFFu;
    __syncthreads();
    if (((tid & 31) == 0) && !wall) atomicAnd(&blkflag, 0u);
    __syncthreads();
    if (tid == 0) {
      if (blkflag == 0u)
        __hip_atomic_fetch_and(&ctrl[2 + i], 0u, __ATOMIC_RELAXED, __HIP_MEMORY_SCOPE_AGENT);
      __hip_atomic_fetch_add(&ctrl[1], 1u, __ATOMIC_RELEASE, __HIP_MEMORY_SCOPE_AGENT);
      const unsigned target = (unsigned)(i + 1) * (unsigned)BLOCKS;
      while (__hip_atomic_load(&ctrl[1], __ATOMIC_ACQUIRE, __HIP_MEMORY_SCOPE_AGENT) < target)
        __builtin_amdgcn_s_sleep(2);
    }
    __syncthreads();
    const unsigned allSmall =
        __hip_atomic_load(&ctrl[2 + i], __ATOMIC_RELAXED, __HIP_MEMORY_SCOPE_AGENT);
    const float g = allSmall ? 0.0f : 1.0f;

    const float* a = m0p + (size_t)i * 9;
    const float* c = m1p + (size_t)i * 9;
    const float w0 = g * (l / sc[s0 + i]);
    const float w1 = g * (l / sc[s1 + i]);
    v0x += w0 * (a[0] * ex + a[1] * ey + a[2] * ez);
    v0y += w0 * (a[3] * ex + a[4] * ey + a[5] * ez);
    v0z += w0 * (a[6] * ex + a[7] * ey + a[8] * ez);
    const float n1x = v1x + w1 * (c[0] * ex + c[1] * ey + c[2] * ez);
    const float n1y = v1y + w1 * (c[3] * ex + c[4] * ey + c[5] * ez);
    const float n1z = v1z + w1 * (c[6] * ex + c[7] * ey + c[8] * ez);

    ob[(size_t)i * 3 + 0] = v0x;
    ob[(size_t)i * 3 + 1] = v0y;
    ob[(size_t)i * 3 + 2] = v0z;
    v0x = n1x; v0y = n1y; v0z = n1z;
  }
  ob[(size_t)(NVERT - 1) * 3 + 0] = v0x;
  ob[(size_t)(NVERT - 1) * 3 + 1] = v0y;
  ob[(size_t)(NVERT - 1) * 3 + 2] = v0z;
}

extern "C" void kernel_launch(void* const* d_in, const int* in_sizes, int n_in,
                              void* d_out, int out_size, void* d_ws, size_t ws_size,
                              hipStream_t stream) {
  (void)in_sizes; (void)n_in; (void)out_size; (void)ws_size;
  const float* verts = (const float*)d_in[0];   // (B, 128, 3)
  const float* nl    = (const float*)d_in[1];   // (B, 127)
  const float* sc    = (const float*)d_in[2];   // (2B, 127)
  const float* ms    = (const float*)d_in[3];   // (2B, 127, 3, 3)
  const float* zm    = (const float*)d_in[4];   // (B, 127)
  float* out = (float*)d_out;                   // (B, 128, 3)

  unsigned* wm   = (unsigned*)d_ws;             // NWAVES*4 words (4 KB)
  unsigned* ctrl = wm + (size_t)NWAVES * 4;     // 129 words

  rods_fast  <<<dim3(BLOCKS), dim3(THREADS), 0, stream>>>(verts, nl, sc, ms, zm, out, wm);
  rods_reduce<<<dim3(1),      dim3(128),     0, stream>>>(wm, ctrl);
  rods_exact <<<dim3(BLOCKS), dim3(THREADS), 0, stream>>>(verts, nl, sc, ms, zm, out, ctrl);
}